// ProteinGraphAttention_64922725646517
// MI455X (gfx1250) — compile-verified
//
#include <hip/hip_runtime.h>
#include <cstddef>

#define NN 50000
#define NE 800000
#define ND 128
#define ED 64
#define NH 8
#define HD 16

static_assert(NN % 16 == 0, "M-tiling assumes NN divisible by 16");
static_assert(NN % 8 == 0,  "layernorm tiling assumes NN divisible by 8");
static_assert((NE * NH) % 256 == 0, "edge grid assumes exact divisibility");

typedef __attribute__((ext_vector_type(2))) float v2f;
typedef __attribute__((ext_vector_type(8))) float v8f;

// ---------------------------------------------------------------------------
// float atomic max via sign-split integer atomics (works for all finite f32)
// ---------------------------------------------------------------------------
__device__ __forceinline__ void atomicMaxF32(float* addr, float val) {
    if (val >= 0.0f) {
        atomicMax((int*)addr, __float_as_int(val));
    } else {
        atomicMin((unsigned int*)addr, __float_as_uint(val));
    }
}

// ---------------------------------------------------------------------------
// 0) init: agg = 0, seg_sum = 0, seg_max = -inf
// ---------------------------------------------------------------------------
__global__ __launch_bounds__(256)
void pga_init_kernel(float* __restrict__ agg, float* __restrict__ seg_sum,
                     float* __restrict__ seg_max) {
    int i = blockIdx.x * 256 + threadIdx.x;
    if (i < NN * ND) agg[i] = 0.0f;
    if (i < NN * NH) {
        seg_sum[i] = 0.0f;
        seg_max[i] = __int_as_float(0xFF800000);  // -inf
    }
}

// ---------------------------------------------------------------------------
// 1) fused Q/K/V projection: [NN,128] @ [128,128] + b, fp32 WMMA 16x16x4.
//    Block = 256 threads = 8 waves; block owns a 16-row strip, wave w owns
//    N-tile w (16 cols). A fragment (b64 load) is reused for all 3 B matrices.
// ---------------------------------------------------------------------------
__global__ __launch_bounds__(256)
void pga_qkv_kernel(const float* __restrict__ nodes,
                    const float* __restrict__ Wq, const float* __restrict__ bq,
                    const float* __restrict__ Wk, const float* __restrict__ bk,
                    const float* __restrict__ Wv, const float* __restrict__ bv,
                    float* __restrict__ q, float* __restrict__ k,
                    float* __restrict__ v) {
    const int lane  = threadIdx.x & 31;
    const int wave  = threadIdx.x >> 5;
    const int n0    = wave * 16;
    const int m0    = blockIdx.x * 16;
    const int mrow  = lane & 15;
    const int khalf = (lane >> 4) << 1;   // 0 or 2 (ISA 16x4 f32 A/B layout)
    const int row   = m0 + mrow;
    const int col   = n0 + mrow;

    const float* arow = nodes + (size_t)row * ND;

    v8f cq = {}; v8f ck = {}; v8f cv = {};
#pragma unroll 4
    for (int k0 = 0; k0 < ND; k0 += 4) {
        const v2f A = *(const v2f*)(arow + k0 + khalf);   // b64, 8B-aligned
        const int b0 = (k0 + khalf) * ND + col;
        v2f Bq, Bk, Bv;
        Bq.x = Wq[b0]; Bq.y = Wq[b0 + ND];
        Bk.x = Wk[b0]; Bk.y = Wk[b0 + ND];
        Bv.x = Wv[b0]; Bv.y = Wv[b0 + ND];
        cq = __builtin_amdgcn_wmma_f32_16x16x4_f32(false, A, false, Bq, (short)0, cq, false, false);
        ck = __builtin_amdgcn_wmma_f32_16x16x4_f32(false, A, false, Bk, (short)0, ck, false, false);
        cv = __builtin_amdgcn_wmma_f32_16x16x4_f32(false, A, false, Bv, (short)0, cv, false, false);
    }

    const float bqc = bq[col], bkc = bk[col], bvc = bv[col];
    const int rbase = m0 + ((lane >> 4) << 3);   // lanes 16-31 hold M = i+8
#pragma unroll
    for (int i = 0; i < 8; ++i) {
        const size_t o = (size_t)(rbase + i) * ND + col;
        q[o] = cq[i] + bqc;
        k[o] = ck[i] + bkc;
        v[o] = cv[i] + bvc;
    }
}

// ---------------------------------------------------------------------------
// 2) conservation MLP per node: sigmoid(relu(c*Wc1+bc1) @ Wc2 + bc2) -> [NN,8]
// ---------------------------------------------------------------------------
__global__ __launch_bounds__(256)
void pga_cons_kernel(const float* __restrict__ cs,
                     const float* __restrict__ Wc1, const float* __restrict__ bc1,
                     const float* __restrict__ Wc2, const float* __restrict__ bc2,
                     float* __restrict__ cons) {
    const int n = blockIdx.x * 256 + threadIdx.x;
    if (n >= NN) return;
    const float c = cs[n];
    float h[32];
#pragma unroll
    for (int j = 0; j < 32; ++j) h[j] = fmaxf(fmaf(c, Wc1[j], bc1[j]), 0.0f);
#pragma unroll
    for (int hh = 0; hh < NH; ++hh) {
        float o = bc2[hh];
#pragma unroll
        for (int j = 0; j < 32; ++j) o = fmaf(h[j], Wc2[j * NH + hh], o);
        cons[(size_t)n * NH + hh] = 1.0f / (1.0f + __expf(-o));
    }
}

// ---------------------------------------------------------------------------
// 3) per-(edge,head) attention score + atomic segment-max (b128 gathers)
// ---------------------------------------------------------------------------
__global__ __launch_bounds__(256)
void pga_score_kernel(const float* __restrict__ q, const float* __restrict__ k,
                      const float* __restrict__ edges, const int* __restrict__ ei,
                      const float* __restrict__ We, const float* __restrict__ be,
                      const float* __restrict__ cons, const float* __restrict__ dist,
                      const float* __restrict__ Wd1, const float* __restrict__ bd1,
                      const float* __restrict__ Wd2, const float* __restrict__ bd2,
                      float* __restrict__ scores, float* __restrict__ seg_max) {
    const int idx = blockIdx.x * 256 + threadIdx.x;
    const int e = idx >> 3;
    const int h = idx & 7;
    const int s = ei[e];
    const int t = ei[NE + e];

    // q[src] . k[tgt] per head / sqrt(16)  (4x b128 loads each, 64B aligned)
    const float4* qr = (const float4*)(q + (size_t)s * ND + h * HD);
    const float4* kr = (const float4*)(k + (size_t)t * ND + h * HD);
    float dot = 0.0f;
#pragma unroll
    for (int i = 0; i < 4; ++i) {
        const float4 a = qr[i];
        const float4 b = kr[i];
        dot = fmaf(a.x, b.x, dot);
        dot = fmaf(a.y, b.y, dot);
        dot = fmaf(a.z, b.z, dot);
        dot = fmaf(a.w, b.w, dot);
    }
    dot *= 0.25f;

    // edge-feature bias (16x b128 loads of the edge row; We is L0-resident)
    float eb = be[h];
    const float4* er = (const float4*)(edges + (size_t)e * ED);
#pragma unroll
    for (int j = 0; j < ED / 4; ++j) {
        const float4 ev = er[j];
        eb = fmaf(ev.x, We[(4 * j + 0) * NH + h], eb);
        eb = fmaf(ev.y, We[(4 * j + 1) * NH + h], eb);
        eb = fmaf(ev.z, We[(4 * j + 2) * NH + h], eb);
        eb = fmaf(ev.w, We[(4 * j + 3) * NH + h], eb);
    }

    float sc = (dot + eb) * cons[(size_t)t * NH + h];

    // distance MLP bias
    const float dd = dist[e];
    float db = bd2[h];
#pragma unroll
    for (int j = 0; j < 16; ++j) {
        const float hv = fmaxf(fmaf(dd, Wd1[j], bd1[j]), 0.0f);
        db = fmaf(hv, Wd2[j * NH + h], db);
    }
    sc += db;

    scores[idx] = sc;
    atomicMaxF32(&seg_max[(size_t)t * NH + h], sc);
}

// ---------------------------------------------------------------------------
// 4) w = exp(s - seg_max[tgt]); scatter-add w into seg_sum and w*v[tgt] into agg
//    (normalization by seg_sum deferred to the output projection).
//    16 independent fire-and-forget f32 atomics per thread pipeline on STOREcnt.
// ---------------------------------------------------------------------------
__global__ __launch_bounds__(256)
void pga_scatter_kernel(const float* __restrict__ scores, const int* __restrict__ ei,
                        const float* __restrict__ seg_max, const float* __restrict__ v,
                        float* __restrict__ seg_sum, float* __restrict__ agg) {
    const int idx = blockIdx.x * 256 + threadIdx.x;
    const int e = idx >> 3;
    const int h = idx & 7;
    const int t = ei[NE + e];
    const float w = __expf(scores[idx] - seg_max[(size_t)t * NH + h]);
    atomicAdd(&seg_sum[(size_t)t * NH + h], w);
    const float4* vr = (const float4*)(v + (size_t)t * ND + h * HD);
    float* ar = agg + (size_t)t * ND + h * HD;
#pragma unroll
    for (int i = 0; i < 4; ++i) {
        const float4 vv = vr[i];
        atomicAdd(&ar[4 * i + 0], w * vv.x);
        atomicAdd(&ar[4 * i + 1], w * vv.y);
        atomicAdd(&ar[4 * i + 2], w * vv.z);
        atomicAdd(&ar[4 * i + 3], w * vv.w);
    }
}

// ---------------------------------------------------------------------------
// 5) output projection: x = nodes + (agg/seg_sum) @ Wo + bo  (fp32 WMMA).
//    Reciprocals of seg_sum are hoisted; head index is compile-time under
//    full unroll because khalf in {0,2} never crosses a 16-boundary.
// ---------------------------------------------------------------------------
__global__ __launch_bounds__(256)
void pga_outproj_kernel(const float* __restrict__ agg, const float* __restrict__ seg_sum,
                        const float* __restrict__ Wo, const float* __restrict__ bo,
                        const float* __restrict__ nodes, float* __restrict__ xout) {
    const int lane  = threadIdx.x & 31;
    const int wave  = threadIdx.x >> 5;
    const int n0    = wave * 16;
    const int m0    = blockIdx.x * 16;
    const int mrow  = lane & 15;
    const int khalf = (lane >> 4) << 1;
    const int row   = m0 + mrow;
    const int col   = n0 + mrow;

    const float* arow = agg + (size_t)row * ND;
    const float* srow = seg_sum + (size_t)row * NH;

    float rs[NH];
#pragma unroll
    for (int hh = 0; hh < NH; ++hh) {
        const float ss = srow[hh];
        rs[hh] = ss > 0.0f ? 1.0f / ss : 0.0f;   // empty segment -> 0 (matches ref)
    }

    v8f c = {};
#pragma unroll
    for (int k0 = 0; k0 < ND; k0 += 4) {
        const int head = k0 >> 4;                 // compile-time constant
        v2f A = *(const v2f*)(arow + k0 + khalf); // b64, 8B-aligned
        A.x *= rs[head];
        A.y *= rs[head];
        const int b0 = (k0 + khalf) * ND + col;
        v2f B;
        B.x = Wo[b0]; B.y = Wo[b0 + ND];
        c = __builtin_amdgcn_wmma_f32_16x16x4_f32(false, A, false, B, (short)0, c, false, false);
    }

    const float boc = bo[col];
    const int rbase = m0 + ((lane >> 4) << 3);
#pragma unroll
    for (int i = 0; i < 8; ++i) {
        const size_t o = (size_t)(rbase + i) * ND + col;
        xout[o] = c[i] + boc + nodes[o];
    }
}

// ---------------------------------------------------------------------------
// 6) layernorm in place: one wave32 per node, shfl_xor reduction, b128 I/O
// ---------------------------------------------------------------------------
__global__ __launch_bounds__(256)
void pga_layernorm_kernel(float* __restrict__ x, const float* __restrict__ gamma,
                          const float* __restrict__ beta) {
    const int wave = threadIdx.x >> 5;
    const int lane = threadIdx.x & 31;
    const int node = blockIdx.x * 8 + wave;
    const size_t base = (size_t)node * ND + lane * 4;

    const float4 vv = *(const float4*)(x + base);
    float s  = vv.x + vv.y + vv.z + vv.w;
    float s2 = fmaf(vv.x, vv.x, fmaf(vv.y, vv.y, fmaf(vv.z, vv.z, vv.w * vv.w)));
#pragma unroll
    for (int off = 16; off >= 1; off >>= 1) {
        s  += __shfl_xor(s, off, 32);
        s2 += __shfl_xor(s2, off, 32);
    }
    const float mean = s * (1.0f / 128.0f);
    const float var  = s2 * (1.0f / 128.0f) - mean * mean;
    const float inv  = rsqrtf(var + 1e-5f);

    const float4 g = *(const float4*)(gamma + lane * 4);
    const float4 b = *(const float4*)(beta + lane * 4);
    float4 o;
    o.x = (vv.x - mean) * inv * g.x + b.x;
    o.y = (vv.y - mean) * inv * g.y + b.y;
    o.z = (vv.z - mean) * inv * g.z + b.z;
    o.w = (vv.w - mean) * inv * g.w + b.w;
    *(float4*)(x + base) = o;
}

// ---------------------------------------------------------------------------
extern "C" void kernel_launch(void* const* d_in, const int* in_sizes, int n_in,
                              void* d_out, int out_size, void* d_ws, size_t ws_size,
                              hipStream_t stream) {
    (void)in_sizes; (void)n_in; (void)out_size; (void)ws_size;

    const float* nodes = (const float*)d_in[0];
    const float* edges = (const float*)d_in[1];
    const int*   ei    = (const int*)  d_in[2];
    const float* cscr  = (const float*)d_in[3];
    const float* dist  = (const float*)d_in[4];
    const float* Wq = (const float*)d_in[5];   const float* bq = (const float*)d_in[6];
    const float* Wk = (const float*)d_in[7];   const float* bk = (const float*)d_in[8];
    const float* Wv = (const float*)d_in[9];   const float* bv = (const float*)d_in[10];
    const float* We = (const float*)d_in[11];  const float* be = (const float*)d_in[12];
    const float* Wc1 = (const float*)d_in[13]; const float* bc1 = (const float*)d_in[14];
    const float* Wc2 = (const float*)d_in[15]; const float* bc2 = (const float*)d_in[16];
    const float* Wd1 = (const float*)d_in[17]; const float* bd1 = (const float*)d_in[18];
    const float* Wd2 = (const float*)d_in[19]; const float* bd2 = (const float*)d_in[20];
    const float* Wo = (const float*)d_in[21];  const float* bo = (const float*)d_in[22];
    const float* gamma = (const float*)d_in[23];
    const float* beta  = (const float*)d_in[24];

    float* xout = (float*)d_out;

    // workspace partition (floats): 4*6.4M + 3*0.4M + 6.4M = ~133 MB
    float* ws     = (float*)d_ws;
    float* q      = ws;                       // NN*ND
    float* k      = q + (size_t)NN * ND;      // NN*ND
    float* v      = k + (size_t)NN * ND;      // NN*ND
    float* agg    = v + (size_t)NN * ND;      // NN*ND
    float* cons   = agg + (size_t)NN * ND;    // NN*NH
    float* smax   = cons + (size_t)NN * NH;   // NN*NH
    float* ssum   = smax + (size_t)NN * NH;   // NN*NH
    float* scores = ssum + (size_t)NN * NH;   // NE*NH

    const int mtiles    = NN / 16;                    // 3125
    const int initBlks  = (NN * ND + 255) / 256;      // covers agg + seg arrays
    const int ehBlks    = (NE * NH) / 256;            // 25000
    const int consBlks  = (NN + 255) / 256;
    const int lnBlks    = NN / 8;

    pga_init_kernel<<<initBlks, 256, 0, stream>>>(agg, ssum, smax);
    pga_qkv_kernel<<<mtiles, 256, 0, stream>>>(nodes, Wq, bq, Wk, bk, Wv, bv, q, k, v);
    pga_cons_kernel<<<consBlks, 256, 0, stream>>>(cscr, Wc1, bc1, Wc2, bc2, cons);
    pga_score_kernel<<<ehBlks, 256, 0, stream>>>(q, k, edges, ei, We, be, cons, dist,
                                                 Wd1, bd1, Wd2, bd2, scores, smax);
    pga_scatter_kernel<<<ehBlks, 256, 0, stream>>>(scores, ei, smax, v, ssum, agg);
    pga_outproj_kernel<<<mtiles, 256, 0, stream>>>(agg, ssum, Wo, bo, nodes, xout);
    pga_layernorm_kernel<<<lnBlks, 256, 0, stream>>>(xout, gamma, beta);
}